// SpikingKenyonCellLayer_42880953483372
// MI455X (gfx1250) — compile-verified
//
#include <hip/hip_runtime.h>
#include <stdint.h>

// ---------------- model constants (from reference) ----------------
#define DT_F          1e-3f
#define V_REST_F      (-0.06f)
#define V_RESET_F     (-0.06f)
#define V_MIN_F       (-0.08f)
#define TAU_REF_F     2e-3f
#define R_M_F         1e9f
#define I_NOISE_STD_F 5e-12f
#define V_NOISE_STD_F 5e-4f
#define TH_JITTER_F   2e-4f
#define INV_TEMP_F    1e3f
#define LOG2E_F       1.4426950f
// sqrt(3)/2^31 : makes (float)(int32)hash * UNI a unit-variance uniform draw
#define UNI_F         8.0651356e-10f

#define BB 64
#define TT 500
#define NN 2000
#define BLOCK 256
#define VEC 2                 // neurons per thread (independent VOPD-pairable chains)
#define DEPTH 16              // async prefetch pipeline depth (ASYNCcnt is 6-bit)

// ---------------- CDNA5 async global->LDS path --------------------
#if defined(__gfx1250__) && __has_builtin(__builtin_amdgcn_global_load_async_to_lds_b32)
#define HAVE_ASYNC 1
#else
#define HAVE_ASYNC 0
#endif

#if defined(__gfx1250__) && __has_builtin(__builtin_amdgcn_global_load_async_to_lds_b64)
#define HAVE_ASYNC64 1
#else
#define HAVE_ASYNC64 0
#endif

#if __has_builtin(__builtin_amdgcn_s_wait_asynccnt)
#define ASYNC_WAIT(n) __builtin_amdgcn_s_wait_asynccnt(n)
#else
#define ASYNC_WAIT(n) asm volatile("s_wait_asynccnt %0" :: "i"(n) : "memory")
#endif

#if HAVE_ASYNC64
#define OPS_PER_STEP 1
#else
#define OPS_PER_STEP 2
#endif

typedef int v2i __attribute__((vector_size(8)));      // 2 x i32 (gcc vector)
typedef __attribute__((address_space(1))) int GInt;   // global int
typedef __attribute__((address_space(3))) int LInt;   // LDS int
typedef __attribute__((address_space(1))) v2i Gv2i;   // global 2xi32
typedef __attribute__((address_space(3))) v2i Lv2i;   // LDS 2xi32
typedef float v2f __attribute__((ext_vector_type(2)));
typedef __attribute__((address_space(3))) v2f Lv2f;   // LDS float2

__device__ __forceinline__ void stage_issue(const float* g, float* l) {
#if HAVE_ASYNC64
    __builtin_amdgcn_global_load_async_to_lds_b64((Gv2i*)g, (Lv2i*)l, 0, 0);
#elif HAVE_ASYNC
    __builtin_amdgcn_global_load_async_to_lds_b32((GInt*)g,       (LInt*)l,       0, 0);
    __builtin_amdgcn_global_load_async_to_lds_b32((GInt*)(g + 1), (LInt*)(l + 1), 0, 0);
#else
    (void)g; (void)l;
#endif
}

// murmur3-style finalizer: good avalanche, ~6 VALU ops
__device__ __forceinline__ uint32_t mix(uint32_t x) {
    x *= 0x7FEB352Du;
    x ^= x >> 15;
    x *= 0x846CA68Bu;
    x ^= x >> 13;
    return x;
}

__global__ __launch_bounds__(BLOCK)
void snn_lif_scan_kernel(const float* __restrict__ current,
                         const float* __restrict__ log_tau_m,
                         const float* __restrict__ v_th,
                         float* __restrict__ out) {
    // 16-deep async staging pipeline: 16 x 256 x 8B = 32 KB of the 320 KB WGP LDS
    __shared__ __align__(16) float stage[DEPTH][BLOCK][VEC];

    const int tid  = threadIdx.x;
    const int gid2 = blockIdx.x * BLOCK + tid;        // pair index, exact fit
    const int b    = gid2 / (NN / VEC);
    const int n    = (gid2 - b * (NN / VEC)) * VEC;   // even -> 8B aligned

    const v2f vth2 = *(const v2f*)(v_th + n);
    // alpha = exp(-DT/tau), tau = exp(log_tau)  =>  division-free:
    const float inv_tau = __builtin_amdgcn_exp2f(-log_tau_m[0] * LOG2E_F);
    const float alpha   = __builtin_amdgcn_exp2f(-DT_F * LOG2E_F * inv_tau);

    const float* __restrict__ src = current + (size_t)b * TT * NN + n;
    float*       __restrict__ dst = out     + (size_t)b * TT * NN + n;

    float    v[VEC], refr[VEC];
    uint32_t c[VEC];
#pragma unroll
    for (int k = 0; k < VEC; ++k) {
        v[k]    = V_RESET_F;
        refr[k] = 0.0f;
        c[k]    = (uint32_t)(b * NN + n + k) * 0x85EBCA6Bu;
    }

#if HAVE_ASYNC
    // prologue: fill the pipeline with steps 0 .. DEPTH-1
#pragma unroll
    for (int p = 0; p < DEPTH; ++p)
        stage_issue(src + (size_t)p * NN, &stage[p][tid][0]);
#endif

    for (int t = 0; t < TT; ++t) {
        v2f Iv;
#if HAVE_ASYNC
        if (t + DEPTH < TT) {
            // keep the pipeline full: issue t+DEPTH, then wait until at most
            // DEPTH ops are outstanding. Async loads complete in order per
            // wave, so asynccnt <= DEPTH*OPS  =>  step t's data is in LDS.
            stage_issue(src + (size_t)(t + DEPTH) * NN,
                        &stage[(t + DEPTH) & (DEPTH - 1)][tid][0]);
            ASYNC_WAIT(DEPTH * OPS_PER_STEP);
        } else {
            ASYNC_WAIT(0);    // tail (last DEPTH steps): drain everything
        }
        asm volatile("" ::: "memory");
        Iv = *(const Lv2f*)&stage[t & (DEPTH - 1)][tid][0];  // ds_load_b64
#else
        Iv = *(const v2f*)(src + (size_t)t * NN);
#endif
        float s_out[VEC];
#pragma unroll
        for (int k = 0; k < VEC; ++k) {   // two independent chains -> VOPD
            const float I = Iv[k];

            // three cheap unit-variance draws (scales folded into sigmas)
            c[k] += 0x9E3779B9u;
            const uint32_t h1 = mix(c[k]);
            const uint32_t h2 = h1 * 0xD2511F53u + 0x9E3779B9u;
            const uint32_t h3 = h2 * 0xCD9E8D57u + 0x85EBCA6Bu;
            const float u_i  = (float)(int32_t)h1;   // ~U(-2^31, 2^31)
            const float u_v  = (float)(int32_t)h2;
            const float u_th = (float)(int32_t)h3;

            // LIF update
            refr[k] = fmaxf(refr[k] - DT_F, 0.0f);
            const bool can_fire = (refr[k] <= 0.0f);

            const float Ieff  = fmaf(u_i, I_NOISE_STD_F * UNI_F, I);
            const float v_inf = fmaf(R_M_F, Ieff, V_REST_F);
            float vv = fmaf(v[k] - v_inf, alpha, v_inf);   // exact exp. integ.
            vv = fmaf(u_v, V_NOISE_STD_F * UNI_F, vv);
            vv = fmaxf(vv, V_MIN_F);
            vv = can_fire ? vv : V_RESET_F;

            const float vc = vv - fmaf(u_th, TH_JITTER_F * UNI_F, vth2[k]);
            // sigmoid(vc/TEMP) = 1 / (1 + exp2(-vc*INV_TEMP*log2e))
            const float e  = __builtin_amdgcn_exp2f(vc * (-INV_TEMP_F * LOG2E_F));
            const float sg = __builtin_amdgcn_rcpf(1.0f + e);
            const float s  = can_fire ? sg : 0.0f;

            const bool spiked = (s > 0.5f);
            v[k]    = spiked ? V_RESET_F : vv;
            refr[k] = spiked ? TAU_REF_F : refr[k];
            s_out[k] = s;
        }

        // 256 MB streaming output: 8B per lane, bypass caches (th:NT)
        const v2f sv = { s_out[0], s_out[1] };
        __builtin_nontemporal_store(sv, (v2f*)(dst + (size_t)t * NN));
    }
}

extern "C" void kernel_launch(void* const* d_in, const int* in_sizes, int n_in,
                              void* d_out, int out_size, void* d_ws, size_t ws_size,
                              hipStream_t stream) {
    (void)in_sizes; (void)n_in; (void)out_size; (void)d_ws; (void)ws_size;
    const float* current   = (const float*)d_in[0];
    const float* log_tau_m = (const float*)d_in[1];
    const float* v_th      = (const float*)d_in[2];
    float* out             = (float*)d_out;

    const int blocks = (BB * NN / VEC) / BLOCK;   // 250, exact
    hipLaunchKernelGGL(snn_lif_scan_kernel, dim3(blocks), dim3(BLOCK), 0, stream,
                       current, log_tau_m, v_th, out);
}